// CAREConv_62199716381201
// MI455X (gfx1250) — compile-verified
//
#include <hip/hip_runtime.h>

#define N_NODES 50000
#define D_IN 128
#define D_OUT 64
#define N_ETYPES 3
#define E_PER_ETYPE 500000
#define P_ETYPE 0.5f

typedef float v2f __attribute__((ext_vector_type(2)));
typedef float v8f __attribute__((ext_vector_type(8)));

// hacc = feat (residual init)
__global__ void k_init(const float* __restrict__ feat, float* __restrict__ hacc) {
    int i = blockIdx.x * blockDim.x + threadIdx.x;
    if (i < N_NODES * D_IN) hacc[i] = feat[i];
}

// zero per-etype scatter accumulator + counts
__global__ void k_zero(float* __restrict__ acc, float* __restrict__ cnt) {
    int i = blockIdx.x * blockDim.x + threadIdx.x;
    if (i < N_NODES * D_IN) acc[i] = 0.0f;
    if (i < N_NODES)        cnt[i] = 0.0f;
}

// one wave (32 lanes) per edge: coalesced 512B gather of feat[src], f32 atomics into acc[dst]
__global__ void k_scatter(const float* __restrict__ feat,
                          const int* __restrict__ src,
                          const int* __restrict__ dst,
                          float* __restrict__ acc,
                          float* __restrict__ cnt) {
    int gid  = blockIdx.x * blockDim.x + threadIdx.x;
    int e    = gid >> 5;
    int lane = gid & 31;
    if (e >= E_PER_ETYPE) return;
    int s = src[e];
    int d = dst[e];
    float4 v = *(const float4*)(feat + (size_t)s * D_IN + lane * 4);
    float* ap = acc + (size_t)d * D_IN + lane * 4;
    atomicAdd(ap + 0, v.x);
    atomicAdd(ap + 1, v.y);
    atomicAdd(ap + 2, v.z);
    atomicAdd(ap + 3, v.w);
    if (lane == 0) atomicAdd(cnt + d, 1.0f);
}

// hacc += 0.5 * tanh(acc / max(cnt,1)); final pass applies the outer tanh
__global__ void k_accum(const float* __restrict__ acc, const float* __restrict__ cnt,
                        float* __restrict__ hacc, int finalpass) {
    int i = blockIdx.x * blockDim.x + threadIdx.x;
    if (i >= N_NODES * D_IN) return;
    int n = i >> 7;                      // D_IN == 128
    float c = fmaxf(cnt[n], 1.0f);
    float v = hacc[i] + P_ETYPE * tanhf(acc[i] / c);
    hacc[i] = finalpass ? tanhf(v) : v;
}

// out[N,64] = h[N,128] @ W[128,64] + b, exact fp32 via V_WMMA_F32_16X16X4_F32.
// Block = 128 threads = 4 waves; block owns a 16-row stripe, wave w owns cols [16w,16w+16).
__global__ void k_gemm(const float* __restrict__ h, const float* __restrict__ W,
                       const float* __restrict__ b, float* __restrict__ out) {
    int lane = threadIdx.x & 31;
    int wave = threadIdx.x >> 5;         // 0..3
    int i0 = blockIdx.x * 16;            // row tile (N_NODES % 16 == 0)
    int j0 = wave * 16;                  // col tile
    int rc = lane & 15;                  // A row / B col
    int kh = (lane >> 4) * 2;            // K base within the 4-slice: 0 or 2

    const float* arow = h + (size_t)(i0 + rc) * D_IN + kh;        // A[row][kh + k0 + {0,1}]
    const float* bcol = W + (size_t)kh * D_OUT + (j0 + rc);       // B[kh + k0 + {0,1}][col]

    v8f c = {};
#pragma unroll
    for (int k0 = 0; k0 < D_IN; k0 += 4) {
        v2f a;  a.x  = arow[k0];
                a.y  = arow[k0 + 1];
        v2f bb; bb.x = bcol[(size_t)k0 * D_OUT];
                bb.y = bcol[(size_t)(k0 + 1) * D_OUT];
        // (neg_a, A, neg_b, B, c_mod, C, reuse_a, reuse_b)
        c = __builtin_amdgcn_wmma_f32_16x16x4_f32(false, a, false, bb,
                                                  (short)0, c, false, false);
    }

    int col = lane & 15;
    float bias = b[j0 + col];
#pragma unroll
    for (int v = 0; v < 8; ++v) {
        int row = v + 8 * (lane >> 4);   // C/D layout: VGPR v, lane half selects M offset
        out[(size_t)(i0 + row) * D_OUT + j0 + col] = c[v] + bias;
    }
}

extern "C" void kernel_launch(void* const* d_in, const int* in_sizes, int n_in,
                              void* d_out, int out_size, void* d_ws, size_t ws_size,
                              hipStream_t stream) {
    const float* feat = (const float*)d_in[0];
    const float* W    = (const float*)d_in[1];
    const float* b    = (const float*)d_in[2];
    const int*   ei   = (const int*)d_in[3];   // [N_ETYPES, 2, E_PER_ETYPE]
    float* out = (float*)d_out;

    float* ws   = (float*)d_ws;
    float* acc  = ws;                                   // N*128 floats
    float* hacc = ws + (size_t)N_NODES * D_IN;          // N*128 floats
    float* cnt  = ws + (size_t)2 * N_NODES * D_IN;      // N floats

    const int nd = N_NODES * D_IN;
    const int eb = (E_PER_ETYPE * 32 + 255) / 256;      // 32 lanes per edge

    k_init<<<(nd + 255) / 256, 256, 0, stream>>>(feat, hacc);

    for (int e = 0; e < N_ETYPES; ++e) {
        const int* src = ei + (size_t)e * 2 * E_PER_ETYPE;
        const int* dst = src + E_PER_ETYPE;
        k_zero<<<(nd + 255) / 256, 256, 0, stream>>>(acc, cnt);
        k_scatter<<<eb, 256, 0, stream>>>(feat, src, dst, acc, cnt);
        k_accum<<<(nd + 255) / 256, 256, 0, stream>>>(acc, cnt, hacc, e == N_ETYPES - 1);
    }

    k_gemm<<<N_NODES / 16, 128, 0, stream>>>(hacc, W, b, out);
}